// S4DKernel_69097433858259
// MI455X (gfx1250) — compile-verified
//
#include <hip/hip_runtime.h>
#include <math.h>

typedef __attribute__((ext_vector_type(2))) float v2f;
typedef __attribute__((ext_vector_type(8))) float v8f;

#define HDIM 1024
#define N2   32
#define LLEN 4096
#define KDIM 64   // contraction length = 2*N2 (real+imag)
#define LDSP 66   // padded LDS row stride

// K[h, 64*i + j] = sum_k aT[k][i] * bM[k][j]
//   aT[n   ][i] =  2*Re(Ct[n] * exp(dtA[n]*64*i))
//   aT[n+32][i] = -2*Im(Ct[n] * exp(dtA[n]*64*i))
//   bM[n   ][j] =    Re(exp(dtA[n]*j))
//   bM[n+32][j] =    Im(exp(dtA[n]*j))
__global__ __launch_bounds__(256) void s4d_wmma_kernel(
    const float* __restrict__ log_dt,
    const float* __restrict__ C_real,
    const float* __restrict__ C_imag,
    const float* __restrict__ log_A_real,
    const float* __restrict__ A_imag,
    float* __restrict__ out)
{
  __shared__ float aT[KDIM][LDSP];
  __shared__ float bM[KDIM][LDSP];
  __shared__ float s_dre[N2], s_dim[N2], s_cre[N2], s_cim[N2];

  const int h = blockIdx.x;
  const int t = threadIdx.x;

  // ---- stage 0: per-n constants dtA, Ct = (C)*(exp(dtA)-1)/A ----
  if (t < N2) {
    const int n = t;
    const float dt  = __expf(log_dt[h]);
    const float Are = -__expf(log_A_real[h * N2 + n]);
    const float Aim = A_imag[h * N2 + n];
    const float dre = Are * dt;
    const float dim = Aim * dt;
    const float er  = __expf(dre);
    float sn, cs;
    __sincosf(dim, &sn, &cs);
    const float numr = er * cs - 1.0f;       // exp(dtA) - 1
    const float numi = er * sn;
    const float inv  = 1.0f / (Are * Are + Aim * Aim);
    const float qr   = (numr * Are + numi * Aim) * inv;   // (exp(dtA)-1)/A
    const float qi   = (numi * Are - numr * Aim) * inv;
    const float Cr   = C_real[h * N2 + n];
    const float Ci   = C_imag[h * N2 + n];
    s_cre[n] = Cr * qr - Ci * qi;
    s_cim[n] = Cr * qi + Ci * qr;
    s_dre[n] = dre;
    s_dim[n] = dim;
  }
  __syncthreads();

  // ---- stage 1: build A (U-side, factor 2 and signs folded) and B (Q-side) ----
  for (int idx = t; idx < N2 * 64; idx += 256) {
    const int n = idx >> 6;
    const int i = idx & 63;
    const float x = (float)(64 * i);
    const float r = __expf(s_dre[n] * x);
    float sn, cs;
    __sincosf(s_dim[n] * x, &sn, &cs);
    const float zr = r * cs, zi = r * sn;
    const float ur = s_cre[n] * zr - s_cim[n] * zi;
    const float ui = s_cre[n] * zi + s_cim[n] * zr;
    aT[n][i]      =  2.0f * ur;
    aT[n + N2][i] = -2.0f * ui;
  }
  for (int idx = t; idx < N2 * 64; idx += 256) {
    const int n = idx >> 6;
    const int j = idx & 63;
    const float x = (float)j;
    const float r = __expf(s_dre[n] * x);
    float sn, cs;
    __sincosf(s_dim[n] * x, &sn, &cs);
    bM[n][j]      = r * cs;
    bM[n + N2][j] = r * sn;
  }
  __syncthreads();

  // ---- stage 2: 64x64x64 GEMM via V_WMMA_F32_16X16X4_F32 ----
  // 16 output tiles of 16x16; 8 waves handle 2 tiles each.
  const int wave  = t >> 5;
  const int lane  = t & 31;
  const int lhalf = lane >> 4;   // 0: lanes 0-15, 1: lanes 16-31
  const int l16   = lane & 15;

  float* const op = out + (size_t)h * LLEN;

  for (int tile = wave * 2; tile < wave * 2 + 2; ++tile) {
    const int ti = tile >> 2;          // i-tile (output row block)
    const int tj = tile & 3;           // j-tile (output col block)
    const int arow = ti * 16 + l16;    // A: lane holds row M, half selects K pair
    const int bcol = tj * 16 + l16;    // B: lane holds col N, half selects K pair
    v8f acc = {};
#pragma unroll
    for (int kk = 0; kk < KDIM; kk += 4) {
      const int kb = kk + lhalf * 2;
      v2f av, bv;
      av.x = aT[kb][arow];
      av.y = aT[kb + 1][arow];
      bv.x = bM[kb][bcol];
      bv.y = bM[kb + 1][bcol];
      acc = __builtin_amdgcn_wmma_f32_16x16x4_f32(
          /*neg_a=*/false, av, /*neg_b=*/false, bv,
          /*c_mod=*/(short)0, acc, /*reuse_a=*/false, /*reuse_b=*/false);
    }
    // C/D layout: VGPR r -> M = lhalf*8 + r, N = l16 (per 16x16 f32 table)
#pragma unroll
    for (int r = 0; r < 8; ++r) {
      const int i = ti * 16 + lhalf * 8 + r;
      op[i * 64 + bcol] = acc[r];
    }
  }
}

extern "C" void kernel_launch(void* const* d_in, const int* in_sizes, int n_in,
                              void* d_out, int out_size, void* d_ws, size_t ws_size,
                              hipStream_t stream) {
  const float* log_dt     = (const float*)d_in[0];
  const float* C_real     = (const float*)d_in[1];
  const float* C_imag     = (const float*)d_in[2];
  const float* log_A_real = (const float*)d_in[3];
  const float* A_imag     = (const float*)d_in[4];
  // d_in[5] is L (=4096), fixed at compile time.
  float* out = (float*)d_out;

  s4d_wmma_kernel<<<HDIM, 256, 0, stream>>>(log_dt, C_real, C_imag,
                                            log_A_real, A_imag, out);
}